// WeekForecastProbARIMA_1151051235843
// MI455X (gfx1250) — compile-verified
//
#include <hip/hip_runtime.h>
#include <hip/hip_bf16.h>

typedef __attribute__((ext_vector_type(2))) float v2f;
typedef __attribute__((ext_vector_type(8))) float v8f;

#define P 48
#define FLEN 168
#define HISTLEN 336
#define NTILE_T 11   // ceil(168/16), last tile half-valid

// ---------------------------------------------------------------------------
// Phase 1: unroll the linear AR recurrence into an affine map.
//   s_t = beta_t * bias + sum_k W[t][k] * init_k
// Thread k (0..47) builds column k of W; thread 48 builds beta.
// ws layout: [ W[168][48] row-major | c0[168] ] where c0[t] = beta_t*bias*sigma + mu
// ---------------------------------------------------------------------------
__global__ void build_weights(const float* __restrict__ phi,
                              const float* __restrict__ bias,
                              const float* __restrict__ mu,
                              const float* __restrict__ sigma,
                              float* __restrict__ ws) {
  __shared__ float S[P + 1][P + FLEN];   // 49 x 216 floats
  const int k = threadIdx.x;
  float ph[P];
#pragma unroll
  for (int j = 0; j < P; ++j) ph[j] = phi[j];

  if (k <= P) {
    // history seed: unit basis vectors for W columns, zeros for beta
    for (int i = 0; i < P; ++i) S[k][i] = (k < P && i == k) ? 1.0f : 0.0f;
    for (int t = 0; t < FLEN; ++t) {
      float acc = (k == P) ? 1.0f : 0.0f;   // bias contributes +1 each step
#pragma unroll
      for (int j = 0; j < P; ++j) acc += ph[j] * S[k][t + j];
      S[k][t + P] = acc;
    }
    if (k < P) {
      for (int t = 0; t < FLEN; ++t) ws[t * P + k] = S[k][P + t];
    } else {
      const float b = bias[0], sg = sigma[0], m = mu[0];
      for (int t = 0; t < FLEN; ++t) ws[FLEN * P + t] = S[P][P + t] * b * sg + m;
    }
  }
}

// ---------------------------------------------------------------------------
// Phase 2: out[b,t] = sum_k (y[b,k]-mu) * W[t][k] + c0[t]  via f32 WMMA.
// One wave32 per 16-batch-row strip; 11 time tiles of 16; K=48 -> 12 WMMAs/tile.
// ---------------------------------------------------------------------------
__global__ void __launch_bounds__(256) forecast_gemm(
    const float* __restrict__ enc_l,
    const float* __restrict__ mu_p,
    const float* __restrict__ sigma_p,
    const float* __restrict__ logsig2_p,
    const float* __restrict__ ws,
    float* __restrict__ out, int Btot) {
  __shared__ float sW[FLEN * P + FLEN];   // 8232 floats = ~33 KB
  for (int i = threadIdx.x; i < FLEN * P + FLEN; i += blockDim.x)
    sW[i] = ws[i];
  __syncthreads();

  const float mu = mu_p[0];
  const float lv = logsig2_p[0] + 2.0f * __logf(sigma_p[0]);

  const int lane = threadIdx.x & 31;
  const int wave = threadIdx.x >> 5;
  const int tile = blockIdx.x * 8 + wave;   // batch tile, 16 rows each
  const int b0 = tile * 16;
  const int m = lane & 15;                  // M for A, N for B/C/D
  const int hi = lane >> 4;

  // ---- Load A fragments: 16x48 f32, ISA 16x4 layout per chunk ----
  // lane(0..15)=row M; VGPR0 holds K=0 (lo half) / K=2 (hi half); VGPR1 K=1/K=3.
  const float* rowp = enc_l + (size_t)(b0 + m) * HISTLEN + (HISTLEN - P);
  v2f a[12];
#pragma unroll
  for (int kk = 0; kk < 12; ++kk) {
    const float2 v = *reinterpret_cast<const float2*>(rowp + 4 * kk + 2 * hi);
    v2f t; t.x = v.x - mu; t.y = v.y - mu;
    a[kk] = t;
  }

  float* out_lv = out + (size_t)Btot * FLEN;

  for (int tt = 0; tt < NTILE_T; ++tt) {
    const int t0 = tt * 16;
    const int tcol = t0 + m;                         // this lane's N / time column
    const int trow = tcol < FLEN ? tcol : FLEN - 1;  // clamp: keep EXEC all-1s for WMMA

    // C preloaded with affine offset c0[t] (same across M rows)
    const float c0v = sW[FLEN * P + trow];
    v8f c = {c0v, c0v, c0v, c0v, c0v, c0v, c0v, c0v};

#pragma unroll
    for (int kk = 0; kk < 12; ++kk) {
      // B fragment (4x16): lane(0..15)=col N; VGPR0 K=0/2, VGPR1 K=1/3.
      const float2 w2 = *reinterpret_cast<const float2*>(&sW[trow * P + 4 * kk + 2 * hi]);
      v2f b; b.x = w2.x; b.y = w2.y;
      c = __builtin_amdgcn_wmma_f32_16x16x4_f32(
          /*neg_a=*/false, a[kk], /*neg_b=*/false, b,
          /*c_mod=*/(short)0, c, /*reuse_a=*/false, /*reuse_b=*/false);
    }

    // D layout: lane L -> N=L&15; VGPR r -> M = r + 8*(L>>4)
    if (tcol < FLEN) {
#pragma unroll
      for (int r = 0; r < 8; ++r) {
        const int M = r + 8 * hi;
        const size_t idx = (size_t)(b0 + M) * FLEN + tcol;
        out[idx] = c[r];
        out_lv[idx] = lv;
      }
    }
  }
}

extern "C" void kernel_launch(void* const* d_in, const int* in_sizes, int n_in,
                              void* d_out, int out_size, void* d_ws, size_t ws_size,
                              hipStream_t stream) {
  // dict order: enc_l, enc_t, enc_w, enc_s, phi, bias, log_sigma2, mu, sigma
  const float* enc_l   = (const float*)d_in[0];
  const float* phi     = (const float*)d_in[4];
  const float* bias    = (const float*)d_in[5];
  const float* logsig2 = (const float*)d_in[6];
  const float* mu      = (const float*)d_in[7];
  const float* sigma   = (const float*)d_in[8];
  float* out = (float*)d_out;
  float* ws  = (float*)d_ws;   // needs (168*48 + 168)*4 = ~33 KB

  const int Btot = in_sizes[0] / HISTLEN;      // 131072
  build_weights<<<1, 64, 0, stream>>>(phi, bias, mu, sigma, ws);

  const int tiles = Btot / 16;                 // 8192 batch tiles
  forecast_gemm<<<tiles / 8, 256, 0, stream>>>(enc_l, mu, sigma, logsig2, ws, out, Btot);
}